// agentGraph_88562225643608
// MI455X (gfx1250) — compile-verified
//
#include <hip/hip_runtime.h>
#include <hip/hip_bf16.h>

// ---- problem constants -------------------------------------------------
#define KN       3072      // N = NUM_EVADER + 1 (graph nodes)
#define KP       13        // NUM_POS
#define KL       4096      // lane table side
#define WPC      96        // words per adjacency column = KN/32
#define NT       192       // 16x16 tiles per output dim = KN/16

typedef __attribute__((ext_vector_type(2))) float v2f;
typedef __attribute__((ext_vector_type(8))) float v8f;

// ---- kernel 1: lane codes + validity ----------------------------------
__global__ __launch_bounds__(256) void lane_kernel(const float* __restrict__ x,
                                                   int* __restrict__ lane,
                                                   float* __restrict__ validf) {
    int i = blockIdx.x * blockDim.x + threadIdx.x;
    if (i >= KN) return;
    const float* b = x + (size_t)i * KP;
    bool valid = b[0] >= 0.0f;
    int code = 0;
#pragma unroll
    for (int k = 0; k < KP - 1; ++k)               // MSB first over bits 0..11
        code = (code << 1) | (b[k] != 0.0f ? 1 : 0);
    lane[i]   = valid ? code : 0;
    validf[i] = valid ? 1.0f : 0.0f;
}

// ---- kernel 2: adjacency bitmask + column degrees -> dinv --------------
// one block per column i; each wave evaluates 32 edges in parallel and packs
// the adjacency word with a single wave32 ballot (v_cmp -> packed mask)
__global__ __launch_bounds__(128) void adj_kernel(const float* __restrict__ topo,
                                                  const int* __restrict__ lane,
                                                  const float* __restrict__ validf,
                                                  unsigned* __restrict__ maskw,
                                                  float* __restrict__ dinv) {
    __shared__ int   s_lane[KN];
    __shared__ float s_valid[KN];
    __shared__ int   s_deg;
    int t = threadIdx.x;
    if (t == 0) s_deg = 0;
    for (int k = t; k < KN; k += blockDim.x) {
        s_lane[k]  = lane[k];
        s_valid[k] = validf[k];
    }
    __syncthreads();

    int  i      = blockIdx.x;
    int  lane_i = s_lane[i];
    bool vi     = s_valid[i] > 0.0f;
    int  ln     = t & 31;
    int  wv     = t >> 5;                          // wave id in block (0..3)
    int  cnt    = 0;

    for (int jw = wv; jw < WPC; jw += 4) {
        int   j  = jw * 32 + ln;                   // one edge per lane
        float tv = topo[(size_t)s_lane[j] * KL + lane_i];   // L2-resident gather
        bool  e  = (vi && (s_valid[j] > 0.0f) && (tv >= 0.0f)) || (j == i);
        unsigned word = (unsigned)__ballot(e);     // wave32: packed adjacency word
        if (ln == 0) {
            maskw[(size_t)i * WPC + jw] = word;
            cnt += __popc(word);
        }
    }
    atomicAdd(&s_deg, cnt);                        // lane0s carry the counts
    __syncthreads();
    if (t == 0) {
        float dg = (float)s_deg;                   // >= 1 (diagonal)
        dinv[i] = dg > 0.0f ? rsqrtf(dg) : 0.0f;
    }
}

// ---- kernel 3: padded W16 [KN][16] and transposed y_t[f][j] ------------
__global__ __launch_bounds__(256) void prep_kernel(const float* __restrict__ x,
                                                   const float* __restrict__ W,
                                                   const float* __restrict__ dinv,
                                                   float* __restrict__ W16,
                                                   float* __restrict__ y_t) {
    int idx = blockIdx.x * blockDim.x + threadIdx.x;
    if (idx >= KN * 16) return;
    int f = idx & 15;
    int n = idx >> 4;
    W16[idx] = (f < KP) ? W[(size_t)n * KP + f] : 0.0f;
    if (f < KP)
        y_t[(size_t)f * KN + n] = dinv[n] * x[(size_t)n * KP + f];
}

// ---- kernel 4: agg16[i][0..15] = dinv[i] * sum_j adj[j,i]*y[j,:] -------
// one wave32 per column; bitmask-gated FMAs, shfl_xor reduction
__global__ __launch_bounds__(256) void agg_kernel(const unsigned* __restrict__ maskw,
                                                  const float* __restrict__ y_t,
                                                  const float* __restrict__ dinv,
                                                  float* __restrict__ agg16) {
    int lane = threadIdx.x & 31;
    int i    = blockIdx.x * (blockDim.x >> 5) + (threadIdx.x >> 5);
    if (i >= KN) return;

    float acc[KP];
#pragma unroll
    for (int f = 0; f < KP; ++f) acc[f] = 0.0f;

    for (int jw = 0; jw < WPC; ++jw) {
        unsigned word = maskw[(size_t)i * WPC + jw];            // uniform load
        float    bf   = (float)((word >> lane) & 1u);
        int      j    = jw * 32 + lane;                          // coalesced per f
#pragma unroll
        for (int f = 0; f < KP; ++f)
            acc[f] += bf * y_t[(size_t)f * KN + j];
    }
#pragma unroll
    for (int f = 0; f < KP; ++f)
#pragma unroll
        for (int off = 16; off > 0; off >>= 1)
            acc[f] += __shfl_xor(acc[f], off, 32);

    if (lane == 0) {
        float s = dinv[i];
#pragma unroll
        for (int f = 0; f < 16; ++f)
            agg16[(size_t)i * 16 + f] = (f < KP) ? s * acc[f] : 0.0f;
    }
}

// ---- kernel 5: out[N,N] = agg16 @ W16^T + bias via V_WMMA_F32_16X16X4_F32
// one wave per 16x16 output tile; K=16 -> 4 chained K=4 WMMA ops (full f32)
__global__ __launch_bounds__(256) void gemm_out_wmma(const float* __restrict__ agg16,
                                                     const float* __restrict__ W16,
                                                     const float* __restrict__ bias,
                                                     float* __restrict__ out) {
    int wave = blockIdx.x * (blockDim.x >> 5) + (threadIdx.x >> 5);
    int lane = threadIdx.x & 31;
    int ti = wave / NT;          // output row tile  (grid sized exactly -> in range)
    int tn = wave % NT;          // output col tile
    int i0 = ti * 16;
    int n0 = tn * 16;

    int m  = lane & 15;          // row (A) / col (B/C/D) owned by this lane
    int kb = (lane >> 4) * 2;    // K sub-offset per ISA f32 16x4 / 4x16 layout

    v8f d = {};
#pragma unroll
    for (int ks = 0; ks < 4; ++ks) {
        int k0 = ks * 4;
        v2f a, b;
        // A: 16x4 f32. V0: lanes0-15 K=k0, lanes16-31 K=k0+2; V1: +1 / +3
        a.x = agg16[(size_t)(i0 + m) * 16 + k0 + kb];
        a.y = agg16[(size_t)(i0 + m) * 16 + k0 + kb + 1];
        // B: 4x16 f32 (K x N), W16 holds W^T rows as [n][k]
        b.x = W16[(size_t)(n0 + m) * 16 + k0 + kb];
        b.y = W16[(size_t)(n0 + m) * 16 + k0 + kb + 1];
        d = __builtin_amdgcn_wmma_f32_16x16x4_f32(
                /*neg_a=*/false, a, /*neg_b=*/false, b,
                /*c_mod=*/(short)0, d, /*reuse_a=*/false, /*reuse_b=*/false);
    }

    float bn = bias[n0 + m];
    int rbase = i0 + 8 * (lane >> 4);              // C/D layout: V r -> row r / r+8
#pragma unroll
    for (int r = 0; r < 8; ++r)
        out[(size_t)(rbase + r) * KN + n0 + m] = d[r] + bn;
}

// ---- launcher ----------------------------------------------------------
extern "C" void kernel_launch(void* const* d_in, const int* in_sizes, int n_in,
                              void* d_out, int out_size, void* d_ws, size_t ws_size,
                              hipStream_t stream) {
    const float* node_feature = (const float*)d_in[0];   // [1,KN,KP]
    const float* topo         = (const float*)d_in[1];   // [1,1,KL,KL]
    const float* gcn_weight   = (const float*)d_in[2];   // [KN,KP]
    const float* gcn_bias     = (const float*)d_in[3];   // [KN]
    float*       out          = (float*)d_out;           // [KN,KN]

    // workspace carve-up (all 256B aligned by construction)
    char* ws = (char*)d_ws;
    size_t o = 0;
    int*      lane   = (int*)(ws + o);      o += (size_t)KN * 4;
    float*    validf = (float*)(ws + o);    o += (size_t)KN * 4;
    float*    dinv   = (float*)(ws + o);    o += (size_t)KN * 4;
    unsigned* maskw  = (unsigned*)(ws + o); o += (size_t)KN * WPC * 4;
    float*    y_t    = (float*)(ws + o);    o += (size_t)16 * KN * 4;
    float*    agg16  = (float*)(ws + o);    o += (size_t)KN * 16 * 4;
    float*    W16    = (float*)(ws + o);    o += (size_t)KN * 16 * 4;
    (void)in_sizes; (void)n_in; (void)out_size; (void)ws_size;

    lane_kernel<<<(KN + 255) / 256, 256, 0, stream>>>(node_feature, lane, validf);

    adj_kernel<<<KN, 128, 0, stream>>>(topo, lane, validf, maskw, dinv);

    prep_kernel<<<(KN * 16 + 255) / 256, 256, 0, stream>>>(node_feature, gcn_weight,
                                                           dinv, W16, y_t);

    agg_kernel<<<KN / 8, 256, 0, stream>>>(maskw, y_t, dinv, agg16);   // 8 waves/blk

    // 192*192 tiles, one wave per tile, 8 waves per block -> exact grid (EXEC all-1s)
    gemm_out_wmma<<<(NT * NT) / 8, 256, 0, stream>>>(agg16, W16, gcn_bias, out);
}